// ScalarMapping_1468878815764
// MI455X (gfx1250) — compile-verified
//
#include <hip/hip_runtime.h>

// Shapes from the reference
#define BB   256
#define CC   512
#define DD   784          // 28*28
#define HALF 392          // DD/2, each half handled by one half-wave lane

typedef float f4  __attribute__((ext_vector_type(4)));
typedef float v2f __attribute__((ext_vector_type(2)));
typedef float v8f __attribute__((ext_vector_type(8)));

// One wave32 per (batch b, 16-channel tile).
// Lane L: channel = c0 + (L & 15), covers half (L >> 4) of the D dimension.
// Final cross-half reduction via one v_wmma_f32_16x16x4_f32:
//   A[m,0] = partial(half0, ch m) + bias[m],  A[m,2] = partial(half1, ch m),
//   A[m,1] = A[m,3] = 0,  B = all-ones  =>  D[m,n] = full dot + bias.
__global__ __launch_bounds__(256) void perchan_dot_wmma(
    const float* __restrict__ x,     // (BB, CC, DD)
    const float* __restrict__ Wm,    // (CC, DD)
    const float* __restrict__ bias,  // (CC,)
    float* __restrict__ out)         // (BB, CC)
{
    const int lane  = threadIdx.x;                           // 0..31 (wave32)
    const int b     = blockIdx.x;                            // batch
    const int ctile = blockIdx.y * blockDim.y + threadIdx.y; // 0..31
    const int c0    = ctile * 16;
    const int m     = lane & 15;     // channel-in-tile == WMMA M row
    const int half  = lane >> 4;     // which half of D
    const int ch    = c0 + m;

    // Rows for channels c0..c0+15 of one batch are contiguous (16*784 floats).
    const f4* __restrict__ xp =
        (const f4*)(x + ((size_t)b * CC + ch) * DD + (size_t)half * HALF);
    const f4* __restrict__ wp =
        (const f4*)(Wm + (size_t)ch * DD + (size_t)half * HALF);

    // Unconditional, coalesced bias load (L2-resident); selected later with
    // v_cndmask so there is no EXEC-predicated load before the WMMA.
    const float bv = bias[ch];

    // 4 independent FMA chains; x is streamed exactly once -> non-temporal.
    // Unroll 7: 98 = 14*7, no remainder; ~7 b128 x-loads clause per body for
    // latency hiding (~8 waves/SIMD need >=4 outstanding lines per lane).
    f4 acc = {0.0f, 0.0f, 0.0f, 0.0f};
#pragma unroll 7
    for (int j = 0; j < HALF / 4; ++j) {
        f4 xv = __builtin_nontemporal_load(xp + j);
        f4 wv = wp[j];
        acc[0] = fmaf(xv[0], wv[0], acc[0]);
        acc[1] = fmaf(xv[1], wv[1], acc[1]);
        acc[2] = fmaf(xv[2], wv[2], acc[2]);
        acc[3] = fmaf(xv[3], wv[3], acc[3]);
    }
    float s = (acc[0] + acc[1]) + (acc[2] + acc[3]);
    s += (half == 0) ? bv : 0.0f;   // branchless, EXEC stays all-ones

    // A-matrix (16x4 f32): lanes 0-15 hold K=0,1 ; lanes 16-31 hold K=2,3.
    v2f a;  a[0] = s;     a[1] = 0.0f;   // unused K slot zeroed
    v2f bm; bm[0] = 1.0f; bm[1] = 1.0f;  // all-ones B => row-sum reduction
    v8f cz = {};                         // C = 0

    v8f d = __builtin_amdgcn_wmma_f32_16x16x4_f32(
        /*neg_a=*/false, a, /*neg_b=*/false, bm,
        /*c_mod=*/(short)0, cz, /*reuse_a=*/false, /*reuse_b=*/false);

    // D layout: VGPR r, lanes 0-15 -> M=r ; lanes 16-31 -> M=r+8. All N equal.
    // Lane 0 stores channels c0..c0+7, lane 16 stores c0+8..c0+15 (contiguous,
    // 64B-aligned -> two b128 stores).
    if (m == 0) {
        f4 lo; lo[0] = d[0]; lo[1] = d[1]; lo[2] = d[2]; lo[3] = d[3];
        f4 hi; hi[0] = d[4]; hi[1] = d[5]; hi[2] = d[6]; hi[3] = d[7];
        float* o = out + (size_t)b * CC + c0 + half * 8;
        *(f4*)(o)     = lo;
        *(f4*)(o + 4) = hi;
    }
}

extern "C" void kernel_launch(void* const* d_in, const int* in_sizes, int n_in,
                              void* d_out, int out_size, void* d_ws, size_t ws_size,
                              hipStream_t stream) {
    (void)in_sizes; (void)n_in; (void)out_size; (void)d_ws; (void)ws_size;
    const float* x    = (const float*)d_in[0]; // (256, 512, 28, 28) f32
    const float* Wm   = (const float*)d_in[1]; // (512, 784) f32
    const float* bias = (const float*)d_in[2]; // (512,) f32
    float* out = (float*)d_out;                // (256, 512, 1) f32

    // grid: 256 batches x 4 blocks of 8 channel-tiles; block: 8 waves of 32.
    dim3 grid(BB, (CC / 16) / 8, 1);
    dim3 block(32, 8, 1);
    perchan_dot_wmma<<<grid, block, 0, stream>>>(x, Wm, bias, out);
}